// model_encdec_19885698580923
// MI455X (gfx1250) — compile-verified
//
#include <hip/hip_runtime.h>
#include <hip/hip_bf16.h>

typedef __attribute__((ext_vector_type(2))) float v2f;
typedef __attribute__((ext_vector_type(8))) float v8f;

#define B_ROWS   1024
#define DSTATE   128
#define M_ROWS   100000
#define TOPK     200
#define SEL_ROWS (B_ROWS * TOPK)   // 204800

// ---------------------------------------------------------------------------
// WMMA helpers: D = A(16xK, LDS row-major) * B + C   (fp32, 16x16x4)
// A-fragment: lane l holds A[l&15][k0 + 2*(l>>4) + v] in VGPR v.
// B row-major  [K,N]: element (k,n) at Bp[k*ldb + n]
// B transposed [N,K]: element (k,n) at Bt[n*ldt + k]   (similarity scoring)
// ---------------------------------------------------------------------------
__device__ __forceinline__ v8f wmma_tile_f32(const float* __restrict__ A, int lda,
                                             const float* __restrict__ Bp, int ldb,
                                             int K, v8f acc) {
    const int lane = threadIdx.x & 31;
    const int m  = lane & 15;
    const int kh = (lane >> 4) << 1;   // 0 or 2
    const int n  = lane & 15;
    for (int k0 = 0; k0 < K; k0 += 4) {
        v2f a, b;
        a.x = A[m * lda + k0 + kh + 0];
        a.y = A[m * lda + k0 + kh + 1];
        b.x = Bp[(size_t)(k0 + kh + 0) * ldb + n];
        b.y = Bp[(size_t)(k0 + kh + 1) * ldb + n];
        acc = __builtin_amdgcn_wmma_f32_16x16x4_f32(
            false, a, false, b, (short)0, acc, false, false);
    }
    return acc;
}

__device__ __forceinline__ v8f wmma_tile_f32_bt(const float* __restrict__ A, int lda,
                                                const float* __restrict__ Bt, size_t ldt,
                                                int K, v8f acc) {
    const int lane = threadIdx.x & 31;
    const int m  = lane & 15;
    const int kh = (lane >> 4) << 1;   // 0 or 2
    const float* __restrict__ brow = Bt + (size_t)(lane & 15) * ldt;  // column n
    #pragma unroll
    for (int k0 = 0; k0 < 128; k0 += 4) {   // K fixed = 128 here
        v2f a, b;
        a.x = A[m * lda + k0 + kh + 0];
        a.y = A[m * lda + k0 + kh + 1];
        b.x = brow[k0 + kh + 0];            // consecutive -> global_load_b64
        b.y = brow[k0 + kh + 1];
        acc = __builtin_amdgcn_wmma_f32_16x16x4_f32(
            false, a, false, b, (short)0, acc, false, false);
    }
    (void)K;
    return acc;
}

// ---------------------------------------------------------------------------
// Kernel 1: state_past = concat(norm_past_state, abs_past_state_social)
// ---------------------------------------------------------------------------
__global__ void concat_state_kernel(const float* __restrict__ a,
                                    const float* __restrict__ b,
                                    float* __restrict__ out) {
    int i = blockIdx.x * blockDim.x + threadIdx.x;  // over 1024*128
    int r = i >> 7, c = i & 127;
    out[i] = (c < 64) ? a[r * 64 + c] : b[r * 64 + (c - 64)];
}

// ---------------------------------------------------------------------------
// Kernel 2: rnorm[m] = 1 / max(||memory_past[m]||, 1e-12)  (one wave per row)
// ---------------------------------------------------------------------------
__global__ void rnorm_kernel(const float* __restrict__ mem,
                             float* __restrict__ rn) {
    int gid  = blockIdx.x * blockDim.x + threadIdx.x;
    int row  = gid >> 5;
    int lane = gid & 31;
    if (row >= M_ROWS) return;
    const float4* mp = (const float4*)(mem + (size_t)row * DSTATE);
    float4 v = mp[lane];
    float s = v.x * v.x + v.y * v.y + v.z * v.z + v.w * v.w;
    #pragma unroll
    for (int off = 16; off > 0; off >>= 1) s += __shfl_xor(s, off, 32);
    if (lane == 0) rn[row] = 1.0f / fmaxf(sqrtf(s), 1e-12f);
}

// ---------------------------------------------------------------------------
// Kernel 3: exact top-200 per row. 16 batch rows per block; similarity
// scores computed on the WMMA pipe (fp32 16x16x4), per-row MSB radix select
// (4 x 8-bit passes) + collection + per-row bitonic sort of (val,~idx) keys.
// memory_past (51.2 MB) is L2-resident, so the 5 scoring sweeps are cheap.
// ---------------------------------------------------------------------------
#define RB       16        // batch rows per block
#define TKT      512       // threads (16 waves)
#define NT       (M_ROWS / 16)   // 6250 column tiles
#define CAND     256       // per-row candidate buffer
#define SST      132       // padded LDS stride for state tile

__global__ void __launch_bounds__(TKT)
topk_kernel(const float* __restrict__ state,  // [B,128]
            const float* __restrict__ mem,    // [M,128]
            const float* __restrict__ rn,     // [M]
            int* __restrict__ idx_out)        // [B,200]
{
    __shared__ float sState[RB * SST];
    __shared__ unsigned hist[RB][256];
    __shared__ unsigned prefixS[RB];
    __shared__ unsigned remkS[RB];
    __shared__ unsigned cntS[RB];
    __shared__ unsigned long long cand[RB][CAND];

    const int b0   = blockIdx.x * RB;
    const int tid  = threadIdx.x;
    const int wave = tid >> 5;
    const int lane = tid & 31;
    const int mh   = (lane >> 4) * 8;   // C-fragment row offset
    const int nloc = lane & 15;         // C-fragment column

    for (int i = tid; i < RB * DSTATE; i += TKT) {
        int r = i >> 7, c = i & 127;
        sState[r * SST + c] = state[(size_t)(b0 + r) * DSTATE + c];
    }
    if (tid < RB) { prefixS[tid] = 0u; remkS[tid] = TOPK; }
    __syncthreads();

    unsigned pmask = 0;
    for (int pass = 0; pass < 4; ++pass) {
        const int shift = 24 - 8 * pass;
        for (int i = tid; i < RB * 256; i += TKT) ((unsigned*)hist)[i] = 0u;
        __syncthreads();

        unsigned pfx[8];
        #pragma unroll
        for (int r = 0; r < 8; ++r) pfx[r] = prefixS[mh + r];

        for (int tile = wave; tile < NT; tile += 16) {
            const int m0 = tile * 16;
            v8f acc = {0.f,0.f,0.f,0.f,0.f,0.f,0.f,0.f};
            acc = wmma_tile_f32_bt(sState, SST, mem + (size_t)m0 * DSTATE,
                                   DSTATE, DSTATE, acc);
            const float scale = rn[m0 + nloc];
            #pragma unroll
            for (int r = 0; r < 8; ++r) {
                unsigned u = __float_as_uint(acc[r] * scale);
                u = (u & 0x80000000u) ? ~u : (u | 0x80000000u);
                if ((u & pmask) == pfx[r])
                    atomicAdd(&hist[mh + r][(u >> shift) & 255], 1u);
            }
        }
        __syncthreads();

        if (tid < RB) {
            unsigned remk = remkS[tid];
            unsigned cum = 0;
            int d = 255;
            for (; d > 0; --d) {
                unsigned h = hist[tid][d];
                if (cum + h >= remk) break;
                cum += h;
            }
            prefixS[tid] |= ((unsigned)d << shift);
            remkS[tid] = remk - cum;
        }
        __syncthreads();
        pmask |= (0xFFu << shift);
    }

    // Collection: gather all (val >= per-row threshold) candidates
    if (tid < RB) cntS[tid] = 0u;
    for (int i = tid; i < RB * CAND; i += TKT) ((unsigned long long*)cand)[i] = 0ULL;
    __syncthreads();

    unsigned thr[8];
    #pragma unroll
    for (int r = 0; r < 8; ++r) thr[r] = prefixS[mh + r];

    for (int tile = wave; tile < NT; tile += 16) {
        const int m0 = tile * 16;
        v8f acc = {0.f,0.f,0.f,0.f,0.f,0.f,0.f,0.f};
        acc = wmma_tile_f32_bt(sState, SST, mem + (size_t)m0 * DSTATE,
                               DSTATE, DSTATE, acc);
        const float scale = rn[m0 + nloc];
        const unsigned mglob = (unsigned)(m0 + nloc);
        #pragma unroll
        for (int r = 0; r < 8; ++r) {
            unsigned u = __float_as_uint(acc[r] * scale);
            u = (u & 0x80000000u) ? ~u : (u | 0x80000000u);
            if (u >= thr[r]) {
                unsigned p = atomicAdd(&cntS[mh + r], 1u);
                if (p < CAND)
                    cand[mh + r][p] = ((unsigned long long)u << 32) |
                                      (unsigned)(0xFFFFFFFFu - mglob);
            }
        }
    }
    __syncthreads();

    // Per-row bitonic sort (descending): 2 rows concurrently, 256 thr/row
    for (int rg = 0; rg < 8; ++rg) {
        const int row = rg * 2 + (tid >> 8);
        unsigned long long* buf = cand[row];
        const int i = tid & 255;
        for (int k = 2; k <= CAND; k <<= 1) {
            for (int j = k >> 1; j > 0; j >>= 1) {
                int l = i ^ j;
                if (l > i) {
                    unsigned long long x = buf[i], y = buf[l];
                    bool desc = ((i & k) == 0);
                    if (desc ? (x < y) : (x > y)) { buf[i] = y; buf[l] = x; }
                }
                __syncthreads();
            }
        }
    }

    for (int i = tid; i < RB * TOPK; i += TKT) {
        const int row = i / TOPK, t = i - row * TOPK;
        unsigned long long kk = cand[row][t];
        idx_out[(size_t)(b0 + row) * TOPK + t] =
            (int)(0xFFFFFFFFu - (unsigned)(kk & 0xFFFFFFFFu));
    }
}

// ---------------------------------------------------------------------------
// Kernel 4: memory_sel[row] = memory_past[idx[row]]  (float4 copies)
// ---------------------------------------------------------------------------
__global__ void gather_kernel(const float* __restrict__ mem,
                              const int* __restrict__ idx,
                              float* __restrict__ out) {
    int i = blockIdx.x * blockDim.x + threadIdx.x;  // over SEL_ROWS*32
    int row = i >> 5, j = i & 31;
    int m = idx[row];
    ((float4*)out)[(size_t)row * 32 + j] =
        ((const float4*)mem)[(size_t)m * 32 + j];
}

// ---------------------------------------------------------------------------
// Fused MLP 128 -> 256 (relu) -> 256 (relu) -> 128, fp32 WMMA 16x16x4.
// Block = 128 threads (4 waves), 16 rows per block.
// ---------------------------------------------------------------------------
#define XS 132   // padded stride for 128-wide tile
#define HS 260   // padded stride for 256-wide tile

__global__ void __launch_bounds__(128)
mlp_kernel(const float* __restrict__ X, int R,
           const float* __restrict__ W1, const float* __restrict__ b1,
           const float* __restrict__ W2, const float* __restrict__ b2,
           const float* __restrict__ W3, const float* __restrict__ b3,
           float* __restrict__ Y) {
    __shared__ float sX [16 * XS];
    __shared__ float sH1[16 * HS];
    __shared__ float sH2[16 * HS];

    const int row0 = blockIdx.x * 16;
    const int tid  = threadIdx.x;
    const int wave = tid >> 5;
    const int lane = tid & 31;
    const int nloc = lane & 15;
    const int mh   = (lane >> 4) * 8;

    for (int i = tid; i < 16 * 128; i += 128) {
        int r = i >> 7, c = i & 127;
        sX[r * XS + c] = X[(size_t)(row0 + r) * 128 + c];
    }
    __syncthreads();

    for (int t = wave; t < 16; t += 4) {
        v8f acc = {0.f,0.f,0.f,0.f,0.f,0.f,0.f,0.f};
        acc = wmma_tile_f32(sX, XS, W1 + t * 16, 256, 128, acc);
        int n = t * 16 + nloc;
        float bias = b1[n];
        #pragma unroll
        for (int r = 0; r < 8; ++r)
            sH1[(mh + r) * HS + n] = fmaxf(acc[r] + bias, 0.f);
    }
    __syncthreads();

    for (int t = wave; t < 16; t += 4) {
        v8f acc = {0.f,0.f,0.f,0.f,0.f,0.f,0.f,0.f};
        acc = wmma_tile_f32(sH1, HS, W2 + t * 16, 256, 256, acc);
        int n = t * 16 + nloc;
        float bias = b2[n];
        #pragma unroll
        for (int r = 0; r < 8; ++r)
            sH2[(mh + r) * HS + n] = fmaxf(acc[r] + bias, 0.f);
    }
    __syncthreads();

    for (int t = wave; t < 8; t += 4) {
        v8f acc = {0.f,0.f,0.f,0.f,0.f,0.f,0.f,0.f};
        acc = wmma_tile_f32(sH2, HS, W3 + t * 16, 128, 256, acc);
        int n = t * 16 + nloc;
        float bias = b3[n];
        #pragma unroll
        for (int r = 0; r < 8; ++r)
            Y[(size_t)(row0 + mh + r) * 128 + n] = acc[r] + bias;
    }
    (void)R;
}

// ---------------------------------------------------------------------------
extern "C" void kernel_launch(void* const* d_in, const int* in_sizes, int n_in,
                              void* d_out, int out_size, void* d_ws, size_t ws_size,
                              hipStream_t stream) {
    const float* norm_past = (const float*)d_in[0];
    const float* abs_past  = (const float*)d_in[1];
    const float* mem       = (const float*)d_in[2];
    const float* qw1 = (const float*)d_in[3];
    const float* qb1 = (const float*)d_in[4];
    const float* qw2 = (const float*)d_in[5];
    const float* qb2 = (const float*)d_in[6];
    const float* qw3 = (const float*)d_in[7];
    const float* qb3 = (const float*)d_in[8];
    const float* mw1 = (const float*)d_in[9];
    const float* mb1 = (const float*)d_in[10];
    const float* mw2 = (const float*)d_in[11];
    const float* mb2 = (const float*)d_in[12];
    const float* mw3 = (const float*)d_in[13];
    const float* mb3 = (const float*)d_in[14];

    float* out        = (float*)d_out;
    float* state_past = out;                                   // [1024,128]
    float* state_sel  = out + (size_t)B_ROWS * DSTATE;         // [1024,128]
    float* mem_sel    = out + (size_t)2 * B_ROWS * DSTATE;     // [1024,200,128]
    float* mem_out    = mem_sel + (size_t)SEL_ROWS * DSTATE;   // [1024,200,128]

    float* rn  = (float*)d_ws;                                          // 400 KB
    int*   idx = (int*)((char*)d_ws + (((size_t)M_ROWS * 4 + 255) & ~(size_t)255));

    concat_state_kernel<<<(B_ROWS * DSTATE) / 256, 256, 0, stream>>>(
        norm_past, abs_past, state_past);

    rnorm_kernel<<<(M_ROWS * 32 + 255) / 256, 256, 0, stream>>>(mem, rn);

    topk_kernel<<<B_ROWS / RB, TKT, 0, stream>>>(state_past, mem, rn, idx);

    gather_kernel<<<(SEL_ROWS * 32) / 256, 256, 0, stream>>>(mem, idx, mem_sel);

    mlp_kernel<<<B_ROWS / 16, 128, 0, stream>>>(
        state_past, B_ROWS, qw1, qb1, qw2, qb2, qw3, qb3, state_sel);
    mlp_kernel<<<SEL_ROWS / 16, 128, 0, stream>>>(
        mem_sel, SEL_ROWS, mw1, mb1, mw2, mb2, mw3, mb3, mem_out);
}